// DirectEnergyStressOutput_18382460027059
// MI455X (gfx1250) — compile-verified
//
#include <hip/hip_runtime.h>

// Sizes from the reference
#define N_ATOMS 200000
#define N_EDGES 5000000
#define N_BATCH 32

typedef __attribute__((ext_vector_type(2))) float v2f;
typedef __attribute__((ext_vector_type(8))) float v8f;

// A^T padded to [5][16]: c_AT[k][j] = A[j][k], j>=6 zero.
// r2 = 1/sqrt(2), r6 = 1/sqrt(6), s23 = sqrt(2/3)
__device__ __constant__ float c_AT[5][16] = {
  {0.f,0.f,0.f,0.f,0.f,0.70710678118654752f, 0,0,0,0,0,0,0,0,0,0},
  {0.f,0.f,0.f,0.70710678118654752f,0.f,0.f, 0,0,0,0,0,0,0,0,0,0},
  {-0.40824829046386307f,0.81649658092772603f,-0.40824829046386307f,0.f,0.f,0.f, 0,0,0,0,0,0,0,0,0,0},
  {0.f,0.f,0.f,0.f,0.70710678118654752f,0.f, 0,0,0,0,0,0,0,0,0,0},
  {-0.70710678118654752f,0.f,0.70710678118654752f,0.f,0.f,0.f, 0,0,0,0,0,0,0,0,0,0},
};

// ---------------------------------------------------------------- zero init
__global__ __launch_bounds__(256) void zero_kernel(float* __restrict__ out, int out_n,
                                                   float* __restrict__ ws, int ws_n) {
  int i = blockIdx.x * blockDim.x + threadIdx.x;
  if (i < out_n) out[i] = 0.f;
  if (i < ws_n)  ws[i]  = 0.f;
}

// ------------------------------------------------- edges: force + virial->batch
__global__ __launch_bounds__(256) void edge_kernel(const float* __restrict__ rij,
                                                   const float* __restrict__ fij,
                                                   const int*   __restrict__ eidx,
                                                   const int*   __restrict__ batch,
                                                   float* __restrict__ force,
                                                   float* __restrict__ ws_stress) {
  __shared__ float s[N_BATCH * 6];
  for (int i = threadIdx.x; i < N_BATCH * 6; i += blockDim.x) s[i] = 0.f;
  __syncthreads();

  const int stride = gridDim.x * blockDim.x;
  for (int e = blockIdx.x * blockDim.x + threadIdx.x; e < N_EDGES; e += stride) {
    // streaming, read-once data: nontemporal so L2 stays hot for force[]
    float fx = __builtin_nontemporal_load(&fij[e * 3 + 0]);
    float fy = __builtin_nontemporal_load(&fij[e * 3 + 1]);
    float fz = __builtin_nontemporal_load(&fij[e * 3 + 2]);
    int src  = __builtin_nontemporal_load(&eidx[e]);
    int dst  = __builtin_nontemporal_load(&eidx[N_EDGES + e]);

    atomicAdd(&force[src * 3 + 0],  fx);
    atomicAdd(&force[src * 3 + 1],  fy);
    atomicAdd(&force[src * 3 + 2],  fz);
    atomicAdd(&force[dst * 3 + 0], -fx);
    atomicAdd(&force[dst * 3 + 1], -fy);
    atomicAdd(&force[dst * 3 + 2], -fz);

    float rx = __builtin_nontemporal_load(&rij[e * 3 + 0]);
    float ry = __builtin_nontemporal_load(&rij[e * 3 + 1]);
    float rz = __builtin_nontemporal_load(&rij[e * 3 + 2]);

    int b = batch[dst];            // hot 800KB table, temporal
    float* sb = &s[b * 6];
    atomicAdd(&sb[0], rx * fx);
    atomicAdd(&sb[1], ry * fy);
    atomicAdd(&sb[2], rz * fz);
    atomicAdd(&sb[3], rx * fy);
    atomicAdd(&sb[4], ry * fz);
    atomicAdd(&sb[5], rz * fx);
  }
  __syncthreads();
  for (int i = threadIdx.x; i < N_BATCH * 6; i += blockDim.x)
    atomicAdd(&ws_stress[i], s[i]);
}

// ------------------------------------------- atoms: per-batch iso/aniso sums
__global__ __launch_bounds__(256) void atom_kernel(const float* __restrict__ iso,
                                                   const float* __restrict__ aniso,
                                                   const int*   __restrict__ batch,
                                                   float* __restrict__ ws_aniso,
                                                   float* __restrict__ ws_iso) {
  __shared__ float sa[N_BATCH * 5];
  __shared__ float si[N_BATCH];
  for (int i = threadIdx.x; i < N_BATCH * 5; i += blockDim.x) sa[i] = 0.f;
  for (int i = threadIdx.x; i < N_BATCH; i += blockDim.x) si[i] = 0.f;
  __syncthreads();

  const int stride = gridDim.x * blockDim.x;
  for (int a = blockIdx.x * blockDim.x + threadIdx.x; a < N_ATOMS; a += stride) {
    int b = batch[a];
    atomicAdd(&si[b], __builtin_nontemporal_load(&iso[a]));
    #pragma unroll
    for (int k = 0; k < 5; ++k)
      atomicAdd(&sa[b * 5 + k], __builtin_nontemporal_load(&aniso[a * 5 + k]));
  }
  __syncthreads();
  for (int i = threadIdx.x; i < N_BATCH * 5; i += blockDim.x) atomicAdd(&ws_aniso[i], sa[i]);
  for (int i = threadIdx.x; i < N_BATCH; i += blockDim.x)     atomicAdd(&ws_iso[i], si[i]);
}

// ------------- finalize: e_tot, stress scale, out_dev, WMMA for S[32,5]@A^T
__global__ __launch_bounds__(32) void finalize_kernel(const float* __restrict__ ws_stress,
                                                      const float* __restrict__ ws_aniso,
                                                      const float* __restrict__ ws_iso,
                                                      const float* __restrict__ volume,
                                                      const float* __restrict__ energy,
                                                      float* __restrict__ out) {
  const int OFF_STRESS = 1 + N_ATOMS * 3;
  const int OFF_ANDEV  = OFF_STRESS + N_BATCH * 6;
  const int OFF_DEV    = OFF_ANDEV + N_BATCH * 6;

  const int lane = threadIdx.x;        // 0..31, one full wave, EXEC all ones
  const int half = lane >> 4;          // 0 or 1
  const int l    = lane & 15;
  const int klo  = half * 2;           // A/B f32 16x16x4 layout: VGPR0=K{0,2}, VGPR1=K{1,3}

  // ---- WMMA: out_andev[32,6] = S[32,5] @ A^T   (K split as 4 + 1) ----
  v2f b1; b1.x = c_AT[klo][l];     b1.y = c_AT[klo + 1][l];
  v2f b2; b2.x = (half == 0) ? c_AT[4][l] : 0.f; b2.y = 0.f;

  // tile 0: rows 0..15
  v2f a1_0; a1_0.x = ws_aniso[l * 5 + klo]; a1_0.y = ws_aniso[l * 5 + klo + 1];
  v2f a2_0; a2_0.x = (half == 0) ? ws_aniso[l * 5 + 4] : 0.f; a2_0.y = 0.f;
  // tile 1: rows 16..31
  const int m1 = 16 + l;
  v2f a1_1; a1_1.x = ws_aniso[m1 * 5 + klo]; a1_1.y = ws_aniso[m1 * 5 + klo + 1];
  v2f a2_1; a2_1.x = (half == 0) ? ws_aniso[m1 * 5 + 4] : 0.f; a2_1.y = 0.f;

  v8f c0 = {};
  c0 = __builtin_amdgcn_wmma_f32_16x16x4_f32(false, a1_0, false, b1, (short)0, c0, false, false);
  c0 = __builtin_amdgcn_wmma_f32_16x16x4_f32(false, a2_0, false, b2, (short)0, c0, false, false);
  v8f c1 = {};
  c1 = __builtin_amdgcn_wmma_f32_16x16x4_f32(false, a1_1, false, b1, (short)0, c1, false, false);
  c1 = __builtin_amdgcn_wmma_f32_16x16x4_f32(false, a2_1, false, b2, (short)0, c1, false, false);

  // C/D layout: VGPR v -> row v (lanes 0-15) / row v+8 (lanes 16-31), col = lane&15
  if (l < 6) {
    #pragma unroll
    for (int v = 0; v < 8; ++v) {
      int row = v + half * 8;
      out[OFF_ANDEV + row * 6 + l]        = c0[v];
      out[OFF_ANDEV + (16 + row) * 6 + l] = c1[v];
    }
  }

  // ---- stress_drv = -s_out / volume, out_dev = [iso,iso,iso,0,0,0] ----
  float vol = volume[lane];
  float inv = -1.0f / vol;
  #pragma unroll
  for (int c = 0; c < 6; ++c)
    out[OFF_STRESS + lane * 6 + c] = ws_stress[lane * 6 + c] * inv;

  float is = ws_iso[lane];
  out[OFF_DEV + lane * 6 + 0] = is;
  out[OFF_DEV + lane * 6 + 1] = is;
  out[OFF_DEV + lane * 6 + 2] = is;
  out[OFF_DEV + lane * 6 + 3] = 0.f;
  out[OFF_DEV + lane * 6 + 4] = 0.f;
  out[OFF_DEV + lane * 6 + 5] = 0.f;

  // ---- e_tot ----
  if (lane == 0) {
    float s = 0.f;
    #pragma unroll
    for (int i = 0; i < N_BATCH; ++i) s += energy[i];
    out[0] = s;
  }
}

extern "C" void kernel_launch(void* const* d_in, const int* in_sizes, int n_in,
                              void* d_out, int out_size, void* d_ws, size_t ws_size,
                              hipStream_t stream) {
  const float* rij    = (const float*)d_in[0];
  const float* fij    = (const float*)d_in[1];
  const float* iso    = (const float*)d_in[2];
  const float* aniso  = (const float*)d_in[3];
  const float* volume = (const float*)d_in[4];
  const float* energy = (const float*)d_in[5];
  const int*   eidx   = (const int*)d_in[6];
  const int*   batch  = (const int*)d_in[7];

  float* out = (float*)d_out;
  float* ws  = (float*)d_ws;
  float* ws_stress = ws;        // 192 floats
  float* ws_aniso  = ws + 192;  // 160 floats
  float* ws_iso    = ws + 352;  // 32 floats

  // 1) zero output (force accumulated via atomics) + workspace accumulators
  int zn = out_size;
  hipLaunchKernelGGL(zero_kernel, dim3((zn + 255) / 256), dim3(256), 0, stream,
                     out, zn, ws, 384);

  // 2) edge scatter: force atomics (L2-resident) + LDS-staged virial->batch
  hipLaunchKernelGGL(edge_kernel, dim3(2048), dim3(256), 0, stream,
                     rij, fij, eidx, batch, out + 1, ws_stress);

  // 3) atom reductions: per-batch aniso[32,5] and iso[32]
  hipLaunchKernelGGL(atom_kernel, dim3(512), dim3(256), 0, stream,
                     iso, aniso, batch, ws_aniso, ws_iso);

  // 4) single-wave finalize with WMMA for the [32,5]x[5,6] matmul
  hipLaunchKernelGGL(finalize_kernel, dim3(1), dim3(32), 0, stream,
                     ws_stress, ws_aniso, ws_iso, volume, energy, out);
}